// MatrixTree_40785009443073
// MI455X (gfx1250) — compile-verified
//
#include <hip/hip_runtime.h>
#include <hip/hip_bf16.h>

// ---------------------------------------------------------------------------
// Matrix-tree marginals: build Laplacian -> blocked in-place Gauss-Jordan
// inverse (32-wide pivot blocks) using V_WMMA_F32_16X16X4_F32 -> finalize.
// B=128 batches of N=512.  Workspace holds the 128 Laplacians (128 MB, fits
// in the 192 MB L2 so the O(N^3) inversion traffic stays on-chip).
// ---------------------------------------------------------------------------

typedef __attribute__((ext_vector_type(2))) float v2f;
typedef __attribute__((ext_vector_type(8))) float v8f;

#define NN     512    // matrix dim
#define KB     32     // pivot block width
#define NSTEP  16     // NN / KB
#define NT16   32     // 16-wide tiles per dim
#define TPB    512    // threads per block in invert kernel (16 waves)
#define NWAV   16

// ---------------------------------------------------------------------------
// Kernel 1: build Laplacian per batch.  One block per batch; thread j owns
// column j, loops over rows i (coalesced across threads).
// ---------------------------------------------------------------------------
__global__ void mt_build_lap(const float* __restrict__ x,
                             const int*   __restrict__ lengths,
                             float*       __restrict__ Lw) {
  const int b   = blockIdx.x;
  const int j   = threadIdx.x;
  const int len = lengths[b];
  const float* xb = x  + (size_t)b * NN * NN;
  float*       Lb = Lw + (size_t)b * NN * NN;

  const bool mj = (j < len);
  float colsum = 0.0f;
  for (int i = 0; i < NN; ++i) {
    float e = __expf(xb[i * NN + j]);
    float A = (i != j && i < len && mj) ? e : 0.0f;
    Lb[i * NN + j] = -A;
    colsum += A;
  }
  if (j == 0) {
    Lb[0] = __expf(xb[0]);                       // L[0][0] = root_0
  } else {
    Lb[j * NN + j] = mj ? colsum : 1.0f;         // diag (identity padding)
    Lb[j]          = mj ? __expf(xb[j * NN + j]) // L[0][j] = root_j
                        : 0.0f;
  }
}

// ---------------------------------------------------------------------------
// Kernel 2: in-place blocked Gauss-Jordan inversion, one workgroup per batch.
// Block recurrence with P = inv(Akk)  (KB = 32):
//   Cnew   = -A[:,K] * P            (column panel, rows != K)
//   A[i,j] += Cnew[i,K] * Rold[K,j] (trailing, i,j != K)
//   Rnew   =  P * Rold              (row panel, cols != K)
//   A[K,K] =  P
// All tile GEMMs are 16x16x32 via 8 chained v_wmma_f32_16x16x4_f32.
// LDS: CP 64KB + RPT 64KB + P/Pneg 8KB + diag scratch ~4KB  (~140 KB).
// ---------------------------------------------------------------------------
__global__ __launch_bounds__(TPB) void mt_invert(float* __restrict__ Lw) {
  __shared__ float CP  [NN * KB];     // new column panel          (64 KB)
  __shared__ float RPT [NN * KB];     // old row panel, transposed [j][k] (64 KB)
  __shared__ float Ssh [KB * 33];     // diagonal block scratch (padded)
  __shared__ float Psh [KB * KB];     // P
  __shared__ float Pneg[KB * KB];     // -P

  float* Lb = Lw + (size_t)blockIdx.x * NN * NN;

  const int tid  = threadIdx.x;
  const int lane = tid & 31;
  const int wv   = tid >> 5;
  const int m    = lane & 15;   // A-row / B-col / D-col index within tile
  const int half = lane >> 4;   // lane half selects K pair / D row half
  const int r0   = tid >> 5;    // diag-block element 0 row (owns 2 elements)
  const int c0   = tid & 31;    // diag-block element col

  for (int kb = 0; kb < NSTEP; ++kb) {
    const int K0 = kb * KB;
    const int pt = 2 * kb;      // first pivot 16-tile index

    // ---- load diagonal block and (old) row panel snapshot into LDS ----
    Ssh[r0 * 33 + c0]        = Lb[(K0 + r0) * NN + K0 + c0];
    Ssh[(r0 + 16) * 33 + c0] = Lb[(K0 + r0 + 16) * NN + K0 + c0];
    for (int idx = tid; idx < KB * NN; idx += TPB) {
      int k = idx >> 9;              // 0..31
      int j = idx & (NN - 1);        // 0..511
      RPT[j * KB + k] = Lb[(K0 + k) * NN + j];
    }
    __syncthreads();

    // ---- invert 32x32 diagonal block in LDS (scalar GJ, no pivoting) ----
    for (int t = 0; t < KB; ++t) {
      const float d  = 1.0f / Ssh[t * 33 + t];
      const float stj  = Ssh[t * 33 + c0];
      const float sit0 = Ssh[r0 * 33 + t];
      const float sij0 = Ssh[r0 * 33 + c0];
      const float sit1 = Ssh[(r0 + 16) * 33 + t];
      const float sij1 = Ssh[(r0 + 16) * 33 + c0];
      __syncthreads();
      float nv0, nv1;
      if (r0 == t)           nv0 = (c0 == t) ? d : stj * d;
      else if (c0 == t)      nv0 = -sit0 * d;
      else                   nv0 = sij0 - sit0 * d * stj;
      if (r0 + 16 == t)      nv1 = (c0 == t) ? d : stj * d;
      else if (c0 == t)      nv1 = -sit1 * d;
      else                   nv1 = sij1 - sit1 * d * stj;
      Ssh[r0 * 33 + c0]        = nv0;
      Ssh[(r0 + 16) * 33 + c0] = nv1;
      __syncthreads();
    }
    {
      float v0 = Ssh[r0 * 33 + c0];
      float v1 = Ssh[(r0 + 16) * 33 + c0];
      Psh [r0 * KB + c0]        = v0;
      Pneg[r0 * KB + c0]        = -v0;
      Psh [(r0 + 16) * KB + c0] = v1;
      Pneg[(r0 + 16) * KB + c0] = -v1;
    }
    __syncthreads();

    // ---- stage 2: column panel  C = A[:,K] * (-P)  (rows != pivot) ----
    // 16x16 output tiles: row-tile ib (skip 2 pivot tiles) x col-half q.
    for (int w = wv; w < NT16 * 2; w += NWAV) {
      const int ib = w >> 1;
      const int q  = w & 1;
      if (ib == pt || ib == pt + 1) continue;
      const float* arow = Lb + (ib * 16 + m) * NN + K0;  // lane's A row, K cols
      v8f acc = {};
#pragma unroll
      for (int kk = 0; kk < 8; ++kk) {
        const int k0 = 4 * kk + 2 * half;
        v2f a, bb;
        a.x  = arow[k0];
        a.y  = arow[k0 + 1];
        bb.x = Pneg[k0 * KB + q * 16 + m];
        bb.y = Pneg[(k0 + 1) * KB + q * 16 + m];
        acc = __builtin_amdgcn_wmma_f32_16x16x4_f32(
            false, a, false, bb, (short)0, acc, false, false);
      }
#pragma unroll
      for (int r = 0; r < 8; ++r) {
        const int row = ib * 16 + r + 8 * half;
        Lb[row * NN + K0 + q * 16 + m] = acc[r];
        CP[row * KB + q * 16 + m]      = acc[r];
      }
    }
    __syncthreads();

    // ---- stage 3+4: trailing update and row panel (cols != pivot) ----
    for (int t = wv; t < NT16 * NT16; t += NWAV) {
      const int ib = t >> 5;
      const int jb = t & 31;
      if (jb == pt || jb == pt + 1) continue;
      const bool pivrow = (ib == pt || ib == pt + 1);
      v8f acc = {};
      if (!pivrow) {
#pragma unroll
        for (int r = 0; r < 8; ++r)
          acc[r] = Lb[(ib * 16 + r + 8 * half) * NN + jb * 16 + m];
      }
      const float* aptr = pivrow ? (Psh + ((ib - pt) * 16 + m) * KB)
                                 : (CP + (ib * 16 + m) * KB);
      const float* bptr = RPT + (jb * 16 + m) * KB;
#pragma unroll
      for (int kk = 0; kk < 8; ++kk) {
        const int k0 = 4 * kk + 2 * half;
        v2f a, bb;
        a.x  = aptr[k0];
        a.y  = aptr[k0 + 1];
        bb.x = bptr[k0];
        bb.y = bptr[k0 + 1];
        acc = __builtin_amdgcn_wmma_f32_16x16x4_f32(
            false, a, false, bb, (short)0, acc, false, false);
      }
#pragma unroll
      for (int r = 0; r < 8; ++r)
        Lb[(ib * 16 + r + 8 * half) * NN + jb * 16 + m] = acc[r];
    }
    // pivot diagonal block := P
    Lb[(K0 + r0) * NN + K0 + c0]        = Psh[r0 * KB + c0];
    Lb[(K0 + r0 + 16) * NN + K0 + c0]   = Psh[(r0 + 16) * KB + c0];
    __syncthreads();
  }
}

// ---------------------------------------------------------------------------
// Kernel 3: out = m2 * (term1 - term2 + eye*roots)
// ---------------------------------------------------------------------------
__global__ void mt_finalize(const float* __restrict__ x,
                            const int*   __restrict__ lengths,
                            const float* __restrict__ inv,
                            float*       __restrict__ out) {
  const int b = blockIdx.y;
  const int i = blockIdx.x;
  const int j = threadIdx.x;
  const int len = lengths[b];
  const float* xb = x   + (size_t)b * NN * NN;
  const float* vb = inv + (size_t)b * NN * NN;

  float res = 0.0f;
  if (i < len && j < len) {
    const float e  = __expf(xb[i * NN + j]);
    const float t1 = (j != 0) ? e * vb[j * NN + j] : 0.0f;   // e * inv[j][j]
    const float t2 = (i != 0) ? e * vb[j * NN + i] : 0.0f;   // e * inv[j][i]
    const float t3 = (i == j) ? e * vb[j * NN] : 0.0f;       // e_jj * inv[j][0]
    res = t1 - t2 + t3;
  }
  out[(size_t)b * NN * NN + i * NN + j] = res;
}

// ---------------------------------------------------------------------------
extern "C" void kernel_launch(void* const* d_in, const int* in_sizes, int n_in,
                              void* d_out, int out_size, void* d_ws, size_t ws_size,
                              hipStream_t stream) {
  const float* x       = (const float*)d_in[0];
  const int*   lengths = (const int*)d_in[1];
  float*       out     = (float*)d_out;
  float*       Lw      = (float*)d_ws;

  const int Bn = in_sizes[1];   // 128 batches

  mt_build_lap<<<dim3(Bn), dim3(NN), 0, stream>>>(x, lengths, Lw);
  mt_invert  <<<dim3(Bn), dim3(TPB), 0, stream>>>(Lw);
  mt_finalize<<<dim3(NN, Bn), dim3(NN), 0, stream>>>(x, lengths, Lw, out);
}